// TemporalOperator_74182675136668
// MI455X (gfx1250) — compile-verified
//
#include <hip/hip_runtime.h>
#include <stdint.h>

// Sliding-window minimum (window = 64) along time, per column.
// out[t,d] = min(signal[max(t-63,0)..t, d]),  T=4096, D=8192, fp32.
// Van Herk-Gil-Werman: out[64b+j] = min(suffix_{b-1}[j], prefix_b[j]).

#define T_DIM 4096
#define D_DIM 8192
#define W     64          // window == time-block length
#define BT    64          // threads per workgroup (2 wave32)
#define VEC   2           // columns per thread (float2 path, b64 transfers)
#define CW    (BT * VEC)  // 128 columns per workgroup

typedef float v2f __attribute__((ext_vector_type(2)));
typedef float v4f __attribute__((ext_vector_type(4)));

// LDS byte-offset of a pointer into a __shared__ array (generic -> AS(3) -> int).
__device__ __forceinline__ uint32_t lds_off_u32(const void* p) {
  return (uint32_t)(size_t)(__attribute__((address_space(3))) const void*)p;
}

__global__ void __launch_bounds__(BT)
sliding_min64_kernel(const float* __restrict__ sig, float* __restrict__ out) {
  // One 16-byte slot per (row j, thread): [cur.x, cur.y, sfx.x, sfx.y].
  // Async b64 fills the low 8B, the suffix scan ds_store_b64's the high 8B,
  // and the forward pass reads the whole slot with one ds_load_b128.
  __shared__ v4f s[W * BT];                       // 64 KB -> 5 WGs/WGP

  const int tid = threadIdx.x;
  const int col = blockIdx.x * CW + tid * VEC;    // 8B-aligned global column pair
  const int tb  = blockIdx.y;                     // time block index
  const long long baseCur = (long long)tb * W * D_DIM + col;

  // ---- 1) Async global->LDS staging of the current 64xCW tile -----------
  // 8 B/lane -> 256 B coalesced per wave per instruction; ASYNCcnt-tracked,
  // no VGPR round trip. Overlaps with the suffix scan below.
  {
    const uint32_t lds0 = lds_off_u32(&s[tid]);
    const float* g = sig + baseCur;
#pragma unroll 16
    for (int j = 0; j < W; ++j) {
      uint64_t ga = (uint64_t)(size_t)(g + (long long)j * D_DIM);
      uint32_t la = lds0 + (uint32_t)(j * BT * 16);
      asm volatile("global_load_async_to_lds_b64 %0, %1, off"
                   :: "v"(la), "v"(ga) : "memory");
    }
  }

  // ---- 2) Backward suffix-min over the PREVIOUS block -> high 8B --------
  // suffix[j] = min(prev[j+1..63]) (empty -> +inf); window for t=64b+j is
  // [64(b-1)+j+1 .. 64b+j]. For tb==0 the virtual previous block is all
  // signal[0], already inside prefix_0[j], so +inf is exact.
  if (tb == 0) {
    const v2f inf2 = {__builtin_inff(), __builtin_inff()};
#pragma unroll 16
    for (int j = 0; j < W; ++j)
      ((v2f*)&s[j * BT + tid])[1] = inf2;
  } else {
    const v2f* prev = (const v2f*)(sig + ((long long)(tb - 1) * W * D_DIM + col));
    v2f sfx = {__builtin_inff(), __builtin_inff()};
#pragma unroll 16
    for (int j = W - 1; j >= 0; --j) {
      ((v2f*)&s[j * BT + tid])[1] = sfx;
      v2f p = prev[(long long)j * (D_DIM / VEC)];   // L2-resident re-read
      sfx.x = fminf(sfx.x, p.x);
      sfx.y = fminf(sfx.y, p.y);
    }
  }

  // ---- 3) Drain our wave's async LDS writes, then forward pass ----------
  // Each thread reads only LDS its own wave's async ops (low 8B) and its own
  // ds_stores (high 8B) wrote, so per-wave waits suffice: no barrier needed.
  asm volatile("s_wait_asynccnt 0x0" ::: "memory");

  v2f run = {__builtin_inff(), __builtin_inff()};
  v2f* outp = (v2f*)(out + baseCur);
#pragma unroll 16
  for (int j = 0; j < W; ++j) {
    v4f slot = s[j * BT + tid];        // one ds_load_b128: cur + sfx together
    run.x = fminf(run.x, slot.x);      // prefix_b[j]
    run.y = fminf(run.y, slot.y);
    v2f v;
    v.x = fminf(run.x, slot.z);        // min(prefix, suffix)
    v.y = fminf(run.y, slot.w);
    // Output is write-once / never re-read: stream past L2 so 128 MB of
    // stores don't evict the L2-resident input.
    __builtin_nontemporal_store(v, outp + (long long)j * (D_DIM / VEC));
  }
}

extern "C" void kernel_launch(void* const* d_in, const int* in_sizes, int n_in,
                              void* d_out, int out_size, void* d_ws, size_t ws_size,
                              hipStream_t stream) {
  const float* sig = (const float*)d_in[0];
  float* out = (float*)d_out;
  dim3 grid(D_DIM / CW, T_DIM / W, 1);   // 64 x 64 = 4096 workgroups
  dim3 block(BT, 1, 1);                  // 64 threads = 2 wave32
  sliding_min64_kernel<<<grid, block, 0, stream>>>(sig, out);
}